// HybridModel_49924699849065
// MI455X (gfx1250) — compile-verified
//
#include <hip/hip_runtime.h>

// ---------------------------------------------------------------------------
// MI455X (gfx1250) implementation.
//
// Phase 1: 128 independent 16-qubit statevector sims. 2^16 f32 = 256KB per
// state -> fits in CDNA5's 320KB-per-WGP LDS (impossible on CDNA4's 64KB/CU).
// One workgroup per circuit, state fully LDS-resident; each MERA block fused
// into one 4x4 unitary applied in a single LDS sweep; embedding computed in
// closed form (product state). Avoids ~2GB of HBM butterfly traffic.
//
// Phase 2: 4-layer MLP head on the (64 x 32) expectation matrix using
// V_WMMA_F32_16X16X4_F32 (f32 WMMA: matches reference f32 numerics).
// Weights staged with GLOBAL_LOAD_ASYNC_TO_LDS_B32 (ASYNCcnt path),
// activation buffers bank-conflict padded.
// ---------------------------------------------------------------------------

#define NQ       16
#define NBLK     29
#define BATCH    64
#define NG       2
#define NSTATE   65536     // 2^16 amplitudes
#define THREADS_Q 512
#define TMLP     128

// MERA block wiring (control / target qubits), from _mera_blocks(16).
// constexpr so the fully-unrolled block loop folds all masks/shifts.
constexpr int kBC[NBLK] = {0, 0,8,4,12, 0,4,8,12, 2,6,10,14,
                           0,2,4,6,8,10,12,14, 1,3,5,7,9,11,13,15};
constexpr int kBT[NBLK] = {8, 4,12,8,0, 2,6,10,14, 4,8,12,0,
                           1,3,5,7,9,11,13,15, 2,4,6,8,10,12,14,0};

typedef __attribute__((ext_vector_type(2))) float v2f;
typedef __attribute__((ext_vector_type(8))) float v8f;

typedef __attribute__((address_space(1))) int g_int;   // global (AS1)
typedef __attribute__((address_space(3))) int l_int;   // LDS (AS3)

// ---------------------------------------------------------------------------
// Async global -> LDS copy (CDNA5 GLOBAL_LOAD_ASYNC_TO_LDS_B32, ASYNCcnt).
// Builtin signature (from compiler diagnostic): param 1 is addrspace(1) int*.
// ---------------------------------------------------------------------------
__device__ __forceinline__ void async_ld_f32(float* lds_dst, const float* gsrc)
{
#if __has_builtin(__builtin_amdgcn_global_load_async_to_lds_b32)
    __builtin_amdgcn_global_load_async_to_lds_b32(
        (g_int*)gsrc, (l_int*)lds_dst, 0, 0);
#else
    // GV addressing: vdst = per-lane LDS byte offset, v[addr] = 64-bit global.
    unsigned lds_off =
        (unsigned)(unsigned long long)(l_int*)lds_dst;
    asm volatile("global_load_async_to_lds_b32 %0, %1, off"
                 :: "v"(lds_off), "v"(gsrc) : "memory");
#endif
}

__device__ __forceinline__ void async_wait_all()
{
#if __has_builtin(__builtin_amdgcn_s_wait_asynccnt)
    __builtin_amdgcn_s_wait_asynccnt(0);
#else
    asm volatile("s_wait_asynccnt 0x0" ::: "memory");
#endif
}

// ---------------------------------------------------------------------------
// Kernel 1: statevector simulation, one workgroup per circuit instance.
// ---------------------------------------------------------------------------
__global__ __launch_bounds__(THREADS_Q)
void qnn_kernel(const float* __restrict__ x,      // (B*G, NQ) angles
                const float* __restrict__ qw,     // (NBLK, 2)
                float* __restrict__ qout)         // (B*G, NQ) expectations
{
    __shared__ float S[NSTATE];                   // 256 KB statevector
    __shared__ float red[THREADS_Q / 32][NQ];     // cross-wave reduction

    const int tid = threadIdx.x;
    const int s   = blockIdx.x;                   // circuit instance 0..127

    // --- per-qubit cos/sin of half-angles --------------------------------
    float cq[NQ], sq[NQ];
#pragma unroll
    for (int w = 0; w < NQ; ++w) {
        float th = 0.5f * x[s * NQ + w];
        cq[w] = cosf(th);
        sq[w] = sinf(th);
    }

    // --- embedding: product state, computed closed-form ------------------
    // flat index bit b corresponds to qubit w = 15 - b (axis 1 is MSB).
    // i = tid (bits 0..8) + 512*j (bits 9..15): factor the 16-term product.
    float plow = 1.0f;
#pragma unroll
    for (int b = 0; b < 9; ++b)
        plow *= ((tid >> b) & 1) ? sq[15 - b] : cq[15 - b];

    for (int j = 0; j < NSTATE / THREADS_Q; ++j) {
        float ph = 1.0f;
#pragma unroll
        for (int b = 0; b < 7; ++b)
            ph *= ((j >> b) & 1) ? sq[6 - b] : cq[6 - b];
        S[j * THREADS_Q + tid] = plow * ph;
    }
    __syncthreads();

    // --- 29 MERA blocks, each fused to a single 4x4 unitary ---------------
    const float hh = 0.70710678118654752f;
#pragma unroll
    for (int k = 0; k < NBLK; ++k) {
        const int c = kBC[k], t = kBT[k];         // compile-time constants
        const float wa = 0.5f * qw[2 * k];
        const float wb = 0.5f * qw[2 * k + 1];
        const float ca = cosf(wa), sa = sinf(wa);
        const float cb = cosf(wb), sb = sinf(wb);

        // U = (RY(a) (x) RY(b)) * CNOT * (H (x) I) in basis |c t>.
        const float RR[4][4] = {
            { ca * cb, -ca * sb, -sa * cb,  sa * sb },
            { ca * sb,  ca * cb, -sa * sb, -sa * cb },
            { sa * cb, -sa * sb,  ca * cb, -ca * sb },
            { sa * sb,  sa * cb,  ca * sb,  ca * cb } };
        const float PM[4][4] = {                   // CNOT * (H (x) I)
            { hh, 0.f,  hh, 0.f },
            { 0.f, hh, 0.f,  hh },
            { 0.f, hh, 0.f, -hh },
            { hh, 0.f, -hh, 0.f } };
        float U[4][4];
#pragma unroll
        for (int i = 0; i < 4; ++i)
#pragma unroll
            for (int j = 0; j < 4; ++j)
                U[i][j] = RR[i][0] * PM[0][j] + RR[i][1] * PM[1][j] +
                          RR[i][2] * PM[2][j] + RR[i][3] * PM[3][j];

        const int pc = 15 - c, pt = 15 - t;
        const int p0 = (pc < pt) ? pc : pt;
        const int p1 = (pc < pt) ? pt : pc;
        const unsigned cbit = 1u << pc, tbit = 1u << pt;

        for (int g = tid; g < NSTATE / 4; g += THREADS_Q) {
            // insert two zero bits at positions p0 < p1 (immediate masks)
            const unsigned low  = (unsigned)g & ((1u << p0) - 1u);
            const unsigned rest = (unsigned)g >> p0;
            const unsigned mid  = rest & ((1u << (p1 - p0 - 1)) - 1u);
            const unsigned high = rest >> (p1 - p0 - 1);
            const unsigned base = low | (mid << (p0 + 1)) | (high << (p1 + 1));

            const float v0 = S[base];
            const float v1 = S[base | tbit];
            const float v2 = S[base | cbit];
            const float v3 = S[base | cbit | tbit];
            S[base]               = U[0][0]*v0 + U[0][1]*v1 + U[0][2]*v2 + U[0][3]*v3;
            S[base | tbit]        = U[1][0]*v0 + U[1][1]*v1 + U[1][2]*v2 + U[1][3]*v3;
            S[base | cbit]        = U[2][0]*v0 + U[2][1]*v1 + U[2][2]*v2 + U[2][3]*v3;
            S[base | cbit | tbit] = U[3][0]*v0 + U[3][1]*v1 + U[3][2]*v2 + U[3][3]*v3;
        }
        __syncthreads();
    }

    // --- Z expectations ----------------------------------------------------
    // i = j*512 + tid: qubits 7..15 read bits 0..8 (constant per thread),
    // qubits 0..6 read bits 9..15 (vary with j). So per amplitude we only
    // accumulate total + 7 signed partials; qubits 7..15 are +- total.
    float acc7[7];
#pragma unroll
    for (int w = 0; w < 7; ++w) acc7[w] = 0.0f;
    float total = 0.0f;

    for (int j = 0; j < NSTATE / THREADS_Q; ++j) {
        const int i    = j * THREADS_Q + tid;
        const float a  = S[i];
        const float pr = a * a;
        total += pr;
#pragma unroll
        for (int w = 0; w < 7; ++w)                 // bit position 15-w = 9+(6-w)
            acc7[w] += ((i >> (15 - w)) & 1) ? -pr : pr;
    }

    float acc[NQ];
#pragma unroll
    for (int w = 0; w < 7; ++w) acc[w] = acc7[w];
#pragma unroll
    for (int w = 7; w < NQ; ++w)                    // sign from tid bits only
        acc[w] = ((tid >> (15 - w)) & 1) ? -total : total;

    // wave32 reduction (warpSize == 32 on gfx1250)
#pragma unroll
    for (int w = 0; w < NQ; ++w)
        for (int off = 16; off > 0; off >>= 1)
            acc[w] += __shfl_xor(acc[w], off, 32);

    const int wave = tid >> 5, lane = tid & 31;
    if (lane == 0) {
#pragma unroll
        for (int w = 0; w < NQ; ++w) red[wave][w] = acc[w];
    }
    __syncthreads();
    if (tid < NQ) {
        float sum = 0.0f;
        for (int wv = 0; wv < THREADS_Q / 32; ++wv) sum += red[wv][tid];
        qout[s * NQ + tid] = sum;   // flat == (b, g*16 + w) layout
    }
}

// ---------------------------------------------------------------------------
// f32 WMMA 16x16 tile GEMM helper: D = A(16xK) * B(Kx16), K % 4 == 0.
// A fragment (16x4): lanes 0-15 hold K=k,k+1 ; lanes 16-31 hold K=k+2,k+3.
// B fragment (4x16): mirrored; row striped across lanes.
// ---------------------------------------------------------------------------
__device__ __forceinline__
v8f wmma_tile(const float* A, int lda, const float* B, int ldb,
              int mbase, int nbase, int K, int lane)
{
    v8f acc = {};
    const int l  = lane & 15;
    const int kh = (lane >> 4) << 1;            // 0 or 2
    for (int k = 0; k < K; k += 4) {
        v2f a, b;
        a.x = A[(mbase + l) * lda + k + kh];
        a.y = A[(mbase + l) * lda + k + kh + 1];
        b.x = B[(k + kh) * ldb + nbase + l];
        b.y = B[(k + kh + 1) * ldb + nbase + l];
        acc = __builtin_amdgcn_wmma_f32_16x16x4_f32(
                  false, a, false, b, (short)0, acc, false, false);
    }
    return acc;
}

// Padded row strides (floats): avoid 64-bank LDS conflicts on A-fragment
// loads, which stride by lda across lanes 0..15.
#define XLD   36
#define H1LD 132
#define H2LD  68
#define H3LD  36

// ---------------------------------------------------------------------------
// Kernel 2: MLP head, one workgroup, 4 waves (one 16-row M-tile each).
// ---------------------------------------------------------------------------
__global__ __launch_bounds__(TMLP)
void mlp_kernel(const float* __restrict__ q,
                const float* __restrict__ w1, const float* __restrict__ b1,
                const float* __restrict__ g1, const float* __restrict__ bt1,
                const float* __restrict__ m1, const float* __restrict__ v1,
                const float* __restrict__ w2, const float* __restrict__ b2,
                const float* __restrict__ g2, const float* __restrict__ bt2,
                const float* __restrict__ m2, const float* __restrict__ v2,
                const float* __restrict__ w3, const float* __restrict__ b3,
                const float* __restrict__ g3, const float* __restrict__ bt3,
                const float* __restrict__ m3, const float* __restrict__ v3,
                const float* __restrict__ w4, const float* __restrict__ b4,
                float* __restrict__ out)
{
    __shared__ float Xs [64 * XLD];
    __shared__ float W1s[32 * 128];
    __shared__ float H1s[64 * H1LD];
    __shared__ float W2s[128 * 64];
    __shared__ float H2s[64 * H2LD];
    __shared__ float W3s[64 * 32];
    __shared__ float H3s[64 * H3LD];
    __shared__ float W4s[32 * 16];               // N padded 9 -> 16
    __shared__ float P1 [5 * 128];               // bias,g,bt,m,v
    __shared__ float P2 [5 * 64];
    __shared__ float P3 [5 * 32];
    __shared__ float B4s[16];

    const int tid = threadIdx.x;

    if (tid == 0) {                              // gfx1250 global_prefetch_b8
        __builtin_prefetch(w1, 0, 0);
        __builtin_prefetch(w2, 0, 0);
        __builtin_prefetch(w3, 0, 0);
    }

    // --- async global->LDS staging (ASYNCcnt path, no VGPR round-trip) ----
    for (int i = tid; i < 64 * 32; i += TMLP)    // q with row padding
        async_ld_f32(&Xs[(i >> 5) * XLD + (i & 31)], &q[i]);
    for (int i = tid; i < 32 * 128; i += TMLP) async_ld_f32(&W1s[i], &w1[i]);
    for (int i = tid; i < 128 * 64; i += TMLP) async_ld_f32(&W2s[i], &w2[i]);
    for (int i = tid; i < 64 * 32;  i += TMLP) async_ld_f32(&W3s[i], &w3[i]);

    // small / transformed tails stay synchronous
    for (int i = tid; i < 32 * 16;  i += TMLP) {
        const int r = i >> 4, cc = i & 15;
        W4s[i] = (cc < 9) ? w4[r * 9 + cc] : 0.0f;
    }
    for (int i = tid; i < 128; i += TMLP) {
        P1[i] = b1[i]; P1[128 + i] = g1[i]; P1[256 + i] = bt1[i];
        P1[384 + i] = m1[i]; P1[512 + i] = v1[i];
    }
    for (int i = tid; i < 64; i += TMLP) {
        P2[i] = b2[i]; P2[64 + i] = g2[i]; P2[128 + i] = bt2[i];
        P2[192 + i] = m2[i]; P2[256 + i] = v2[i];
    }
    for (int i = tid; i < 32; i += TMLP) {
        P3[i] = b3[i]; P3[32 + i] = g3[i]; P3[64 + i] = bt3[i];
        P3[96 + i] = m3[i]; P3[128 + i] = v3[i];
    }
    if (tid < 16) B4s[tid] = (tid < 9) ? b4[tid] : 0.0f;

    async_wait_all();
    __syncthreads();

    const int lane  = tid & 31;
    const int mbase = (tid >> 5) * 16;           // 4 waves -> M tiles 0..3
    const int col16 = lane & 15;
    const int rhi   = (lane >> 4) << 3;          // +8 rows for lanes 16-31

    // Layer 1: (64x32)@(32x128), BN+ReLU
    for (int nt = 0; nt < 8; ++nt) {
        v8f d = wmma_tile(Xs, XLD, W1s, 128, mbase, nt * 16, 32, lane);
        const int col = nt * 16 + col16;
        const float sc = P1[128 + col] * rsqrtf(P1[512 + col] + 1e-5f);
#pragma unroll
        for (int r = 0; r < 8; ++r) {
            float val = d[r] + P1[col];
            val = sc * (val - P1[384 + col]) + P1[256 + col];
            H1s[(mbase + r + rhi) * H1LD + col] = fmaxf(val, 0.0f);
        }
    }
    __syncthreads();

    // Layer 2: (64x128)@(128x64), BN+ReLU
    for (int nt = 0; nt < 4; ++nt) {
        v8f d = wmma_tile(H1s, H1LD, W2s, 64, mbase, nt * 16, 128, lane);
        const int col = nt * 16 + col16;
        const float sc = P2[64 + col] * rsqrtf(P2[256 + col] + 1e-5f);
#pragma unroll
        for (int r = 0; r < 8; ++r) {
            float val = d[r] + P2[col];
            val = sc * (val - P2[192 + col]) + P2[128 + col];
            H2s[(mbase + r + rhi) * H2LD + col] = fmaxf(val, 0.0f);
        }
    }
    __syncthreads();

    // Layer 3: (64x64)@(64x32), BN+ReLU
    for (int nt = 0; nt < 2; ++nt) {
        v8f d = wmma_tile(H2s, H2LD, W3s, 32, mbase, nt * 16, 64, lane);
        const int col = nt * 16 + col16;
        const float sc = P3[32 + col] * rsqrtf(P3[128 + col] + 1e-5f);
#pragma unroll
        for (int r = 0; r < 8; ++r) {
            float val = d[r] + P3[col];
            val = sc * (val - P3[96 + col]) + P3[64 + col];
            H3s[(mbase + r + rhi) * H3LD + col] = fmaxf(val, 0.0f);
        }
    }
    __syncthreads();

    // Layer 4: (64x32)@(32x9) + b4, N padded to 16
    {
        v8f d = wmma_tile(H3s, H3LD, W4s, 16, mbase, 0, 32, lane);
#pragma unroll
        for (int r = 0; r < 8; ++r) {
            if (col16 < 9)
                out[(mbase + r + rhi) * 9 + col16] = d[r] + B4s[col16];
        }
    }
}

// ---------------------------------------------------------------------------
extern "C" void kernel_launch(void* const* d_in, const int* in_sizes, int n_in,
                              void* d_out, int out_size, void* d_ws, size_t ws_size,
                              hipStream_t stream)
{
    const float* x   = (const float*)d_in[0];
    const float* qw  = (const float*)d_in[1];
    const float* w1  = (const float*)d_in[2];
    const float* b1  = (const float*)d_in[3];
    const float* g1  = (const float*)d_in[4];
    const float* bt1 = (const float*)d_in[5];
    const float* m1  = (const float*)d_in[6];
    const float* v1  = (const float*)d_in[7];
    const float* w2  = (const float*)d_in[8];
    const float* b2  = (const float*)d_in[9];
    const float* g2  = (const float*)d_in[10];
    const float* bt2 = (const float*)d_in[11];
    const float* m2  = (const float*)d_in[12];
    const float* v2  = (const float*)d_in[13];
    const float* w3  = (const float*)d_in[14];
    const float* b3  = (const float*)d_in[15];
    const float* g3  = (const float*)d_in[16];
    const float* bt3 = (const float*)d_in[17];
    const float* m3  = (const float*)d_in[18];
    const float* v3  = (const float*)d_in[19];
    const float* w4  = (const float*)d_in[20];
    const float* b4  = (const float*)d_in[21];

    float* qbuf = (float*)d_ws;                  // (128, 16) expectations

    qnn_kernel<<<BATCH * NG, THREADS_Q, 0, stream>>>(x, qw, qbuf);
    mlp_kernel<<<1, TMLP, 0, stream>>>(qbuf,
        w1, b1, g1, bt1, m1, v1,
        w2, b2, g2, bt2, m2, v2,
        w3, b3, g3, bt3, m3, v3,
        w4, b4, (float*)d_out);
}